// CompositeValueNoise_43662637531393
// MI455X (gfx1250) — compile-verified
//
#include <hip/hip_runtime.h>
#include <hip/hip_bf16.h>

// CompositeValueNoise for MI455X (gfx1250, wave32).
// Gather-bound workload: 4 octaves of trilinear value noise over L2-resident
// grids. WMMA intentionally NOT used (per-point 8-term dot products have no
// shared-B structure; a 16x16 WMMA would waste 15/16 of its MACs). CDNA5 paths
// exercised instead: async global->LDS DMA (ASYNCcnt) for the coarse grid,
// ds_load_b128 LDS gathers, b128 global gathers, global_prefetch_b8, and
// non-temporal hints to keep the grids resident in the 192MB L2 while the
// x/out streams pass through once.

namespace {

typedef float v4f __attribute__((ext_vector_type(4)));   // native vector: OK for NT builtins

constexpr int NF = 4;                           // fields per grid point
constexpr int V16_FLOATS = 17 * 17 * 17 * NF;   // 19652 floats = 78,608 B (fits in 320KB LDS)
constexpr int V16_VEC4   = V16_FLOATS / 4;      // 4913 b128 transfers

__device__ __forceinline__ v4f ld4(const float* p) {
    return *reinterpret_cast<const v4f*>(p);
}

__device__ __forceinline__ v4f lerp4(v4f a, v4f b, float w) {
    v4f r;
    r.x = fmaf(w, b.x - a.x, a.x);
    r.y = fmaf(w, b.y - a.y, a.y);
    r.z = fmaf(w, b.z - a.z, a.z);
    r.w = fmaf(w, b.w - a.w, a.w);
    return r;
}

// One octave: smoothstep-weighted trilinear interpolation of V at (px,py,pz)*RES.
// RES is a power of two, so px*RES is exact and fmod(x*res,res)==x*res for x in [0,1).
template <int RES>
__device__ __forceinline__ void accum_level(const float* __restrict__ V,
                                            float px, float py, float pz,
                                            float mult, v4f& acc) {
    constexpr int R1 = RES + 1;
    constexpr int sY = R1 * NF;        // stride of dim1 in floats
    constexpr int sX = R1 * R1 * NF;   // stride of dim0 in floats
    const float rf = (float)RES;

    float sx = px * rf, sy = py * rf, sz = pz * rf;
    float fx = floorf(sx), fy = floorf(sy), fz = floorf(sz);
    int   ix = (int)fx,    iy = (int)fy,    iz = (int)fz;
    float tx = sx - fx,    ty = sy - fy,    tz = sz - fz;
    float wx = tx * tx * (3.0f - 2.0f * tx);
    float wy = ty * ty * (3.0f - 2.0f * ty);
    float wz = tz * tz * (3.0f - 2.0f * tz);

    const float* b00 = V + (ix * sX + iy * sY + iz * NF); // (ix,   iy,   iz)
    const float* b01 = b00 + sY;                          // (ix,   iy+1, iz)
    const float* b10 = b00 + sX;                          // (ix+1, iy,   iz)
    const float* b11 = b10 + sY;                          // (ix+1, iy+1, iz)

    // z and z+1 corners are contiguous 32B: two aligned float4 loads each.
    v4f v000 = ld4(b00), v001 = ld4(b00 + NF);
    v4f v010 = ld4(b01), v011 = ld4(b01 + NF);
    v4f v100 = ld4(b10), v101 = ld4(b10 + NF);
    v4f v110 = ld4(b11), v111 = ld4(b11 + NF);

    // Match reference lerp order: dim0 (x) first, then dim1 (y), then dim2 (z).
    v4f c00 = lerp4(v000, v100, wx);
    v4f c01 = lerp4(v001, v101, wx);
    v4f c10 = lerp4(v010, v110, wx);
    v4f c11 = lerp4(v011, v111, wx);
    v4f d0  = lerp4(c00, c10, wy);
    v4f d1  = lerp4(c01, c11, wy);
    v4f r   = lerp4(d0, d1, wz);

    acc.x = fmaf(r.x, mult, acc.x);
    acc.y = fmaf(r.y, mult, acc.y);
    acc.z = fmaf(r.z, mult, acc.z);
    acc.w = fmaf(r.w, mult, acc.w);
}

__device__ __forceinline__ v4f do_point(const float* __restrict__ x,
                                        const float* __restrict__ sV16,
                                        const float* __restrict__ V32,
                                        const float* __restrict__ V64,
                                        const float* __restrict__ V128,
                                        int p) {
    const float* xp = x + 3 * (size_t)p;
    // x is single-use streaming data: non-temporal loads (TH=NT).
    float px = __builtin_nontemporal_load(xp + 0);
    float py = __builtin_nontemporal_load(xp + 1);
    float pz = __builtin_nontemporal_load(xp + 2);

    v4f acc = {0.0f, 0.0f, 0.0f, 0.0f};
    accum_level<16 >(sV16, px, py, pz, 1.000f, acc);  // LDS gathers (ds_load_b128)
    accum_level<32 >(V32,  px, py, pz, 0.500f, acc);  // L2-resident gathers
    accum_level<64 >(V64,  px, py, pz, 0.250f, acc);
    accum_level<128>(V128, px, py, pz, 0.125f, acc);
    return acc;
}

} // namespace

__global__ __launch_bounds__(256)
void composite_value_noise_kernel(const float* __restrict__ x,
                                  const float* __restrict__ V16,
                                  const float* __restrict__ V32,
                                  const float* __restrict__ V64,
                                  const float* __restrict__ V128,
                                  float* __restrict__ out,
                                  int n) {
    // --- Stage V16 (78.6 KB) into LDS with CDNA5 async global->LDS DMA. ---
    __shared__ __align__(16) float sV16[V16_FLOATS];
    {
        for (int i = threadIdx.x; i < V16_VEC4; i += blockDim.x) {
            const float* gsrc = V16 + i * 4;
            // Low 32 bits of a flat shared-aperture address are the LDS byte offset.
            unsigned lds_off = (unsigned)(uintptr_t)(&sV16[i * 4]);
            asm volatile("global_load_async_to_lds_b128 %0, %1, off"
                         :
                         : "v"(lds_off), "v"(gsrc)
                         : "memory");
        }
        asm volatile("s_wait_asynccnt 0" ::: "memory");
        __syncthreads();
    }

    v4f* out4 = reinterpret_cast<v4f*>(out);
    const int stride = gridDim.x * blockDim.x;
    int p = blockIdx.x * blockDim.x + threadIdx.x;

    // Process two points per iteration: ~64 independent b128 gathers in flight
    // per thread to hide L2 gather latency (occupancy is LDS-capped at ~4
    // blocks/WGP, so per-thread MLP carries the latency hiding).
    for (; p + stride < n; p += 2 * stride) {
        int q = p + stride;
        __builtin_prefetch(&x[3 * (size_t)(p + 2 * stride)], 0, 0);

        v4f accA = do_point(x, sV16, V32, V64, V128, p);
        v4f accB = do_point(x, sV16, V32, V64, V128, q);

        // out is single-use streaming data: non-temporal stores (TH=NT).
        __builtin_nontemporal_store(accA, out4 + p);
        __builtin_nontemporal_store(accB, out4 + q);
    }
    if (p < n) {
        v4f accA = do_point(x, sV16, V32, V64, V128, p);
        __builtin_nontemporal_store(accA, out4 + p);
    }
}

extern "C" void kernel_launch(void* const* d_in, const int* in_sizes, int n_in,
                              void* d_out, int out_size, void* d_ws, size_t ws_size,
                              hipStream_t stream) {
    const float* x    = (const float*)d_in[0];
    const float* V16  = (const float*)d_in[1];
    const float* V32  = (const float*)d_in[2];
    const float* V64  = (const float*)d_in[3];
    const float* V128 = (const float*)d_in[4];
    float* out = (float*)d_out;

    const int n = in_sizes[0] / 3;  // number of points

    // Persistent-style grid: amortize the 78KB V16 LDS staging over ~2K points
    // per block (1024 blocks x 256 threads = 8 waves/block, grid-stride loop).
    const int threads = 256;
    int blocks = 1024;
    const int max_blocks = (n + threads - 1) / threads;
    if (blocks > max_blocks) blocks = max_blocks > 0 ? max_blocks : 1;

    composite_value_noise_kernel<<<blocks, threads, 0, stream>>>(
        x, V16, V32, V64, V128, out, n);
}